// SEALModel_4355096838710
// MI455X (gfx1250) — compile-verified
//
#include <hip/hip_runtime.h>

#define G_   8192
#define NPG  64
#define NN   (G_ * NPG)      // 524288 nodes
#define EE   2097152         // edges
#define INC  128
#define HH   32
#define KK   30
#define EPS_ 1e-5f

typedef __attribute__((ext_vector_type(2))) float v2f;
typedef __attribute__((ext_vector_type(8))) float v8f;

// ---------------------------------------------------------------------------
// fp32 WMMA GEMM, A read exactly once, B staged in LDS per workgroup:
//   C[M x NT*16] = A[M x Kd] * Bswz[Kd x NT*16]   (+ optional bias/ReLU)
// One wave owns a 16-row stripe of A, accumulating all NT column tiles in
// registers.  B is PRE-SWIZZLED into fragment order
//     Bswz[((r>>1)*LDB + col)*2 + (r&1)] = B[r][col]
// so that the pair {B[k+kh][col], B[k+kh+1][col]} needed by a WMMA B-operand
// is 8 contiguous, 8B-aligned bytes -> one ds_load_b64 per fragment, no
// register-pairing moves.  Staged into LDS in KROWS-row chunks (Kd % KROWS
// == 0, KROWS % 4 == 0); the K-chunk is fully unrolled (immediate DS offsets),
// then NT back-to-back V_WMMA_F32_16X16X4_F32 issue per A fragment.
// A-frag (ISA 7.12.2, 32-bit A 16x4): lanes 0-15 hold K={0,1}, lanes 16-31
// hold K={2,3}.  C/D: VGPR v -> row v + 8*(lane>=16), col lane&15.
// ---------------------------------------------------------------------------
template<int NT, int KROWS>
__global__ __launch_bounds__(128)
void gemm_wmma_f32(const float* __restrict__ A, int lda,
                   const float* __restrict__ Bswz,   // swizzled, ldb = NT*16
                   float* __restrict__ C,            // ldc = NT*16
                   int M, int Kd,
                   const float* __restrict__ bias, int relu)
{
    constexpr int LDB = NT * 16;
    __shared__ float Bs[KROWS * LDB];

    const int  wave   = blockIdx.x * (blockDim.x >> 5) + (threadIdx.x >> 5);
    const int  lane   = threadIdx.x & 31;
    const bool active = wave < (M >> 4);              // wave-uniform

    const int row0 = wave << 4;
    const int lrow = lane & 15;
    const int kh   = (lane >> 4) << 1;                // 0 for lanes 0-15, 2 for 16-31
    const int h    = lane >> 4;                       // kh/2

    v8f acc[NT];
#pragma unroll
    for (int t = 0; t < NT; ++t) acc[t] = (v8f){};

    const float* ap = A + (size_t)(row0 + lrow) * lda + kh;

    for (int k0 = 0; k0 < Kd; k0 += KROWS) {
        __syncthreads();                              // protect previous chunk
        for (int i = threadIdx.x; i < KROWS * LDB; i += 128)
            Bs[i] = Bswz[(size_t)k0 * LDB + i];       // chunk-local permutation
        __syncthreads();

        if (active) {
            const float* bp = &Bs[(h * LDB + lrow) * 2];
#pragma unroll
            for (int kk = 0; kk < KROWS; kk += 4) {
                v2f a = *(const v2f*)ap;              // A[row][k+kh .. k+kh+1]
                v2f bf[NT];
#pragma unroll
                for (int t = 0; t < NT; ++t)          // one ds_load_b64 each
                    bf[t] = *(const v2f*)(bp + t * 32);
#pragma unroll
                for (int t = 0; t < NT; ++t)          // NT back-to-back WMMAs
                    acc[t] = __builtin_amdgcn_wmma_f32_16x16x4_f32(
                                 false, a, false, bf[t], (short)0, acc[t],
                                 false, false);
                ap += 4;
                bp += 4 * LDB;                        // next k4 group
            }
        }
    }

    if (!active) return;
    const int orow = row0 + ((lane >> 4) << 3);
#pragma unroll
    for (int t = 0; t < NT; ++t) {
        const int ocol = t * 16 + lrow;
        const float bsc = bias ? bias[ocol] : 0.0f;
#pragma unroll
        for (int v = 0; v < 8; ++v) {
            float val = acc[t][v] + bsc;
            if (relu) val = fmaxf(val, 0.0f);
            C[(size_t)(orow + v) * LDB + ocol] = val;
        }
    }
}

// Pack a (K x 32) row-major weight block into columns [c0, c0+32) of the
// swizzled image Wswz (K x ldw): Wswz[((r>>1)*ldw + c0+c)*2 + (r&1)] = w[r][c]
__global__ void pack_cols_swz(const float* __restrict__ w, int K,
                              float* __restrict__ Wswz, int ldw, int c0)
{
    int t = blockIdx.x * blockDim.x + threadIdx.x;
    if (t >= K * 32) return;
    int r = t >> 5, c = t & 31;
    Wswz[((size_t)(r >> 1) * ldw + c0 + c) * 2 + (r & 1)] = w[t];
}

// ---------------------------------------------------------------------------
__global__ void zero_kernel(float* __restrict__ p, size_t n)
{
    size_t i = (size_t)blockIdx.x * blockDim.x + threadIdx.x;
    if (i < n) p[i] = 0.0f;
}

__global__ void degree_kernel(const int* __restrict__ dst,
                              float* __restrict__ deg, int nedges)
{
    int e = blockIdx.x * blockDim.x + threadIdx.x;
    if (e < nedges) atomicAdd(&deg[dst[e]], 1.0f);
}

// One wave32 per edge: lane c adds P[src][c] into agg[dst][c] (coalesced).
__global__ void scatter_add32(const float* __restrict__ P, int ldp,
                              const int* __restrict__ src,
                              const int* __restrict__ dst,
                              float* __restrict__ agg, int nedges)
{
    int t = blockIdx.x * blockDim.x + threadIdx.x;
    int e = t >> 5;
    int c = t & 31;
    if (e >= nedges) return;
    int s = src[e], d = dst[e];
    atomicAdd(&agg[(size_t)d * HH + c], P[(size_t)s * ldp + c]);
}

// mean-div + lin_l bias + lin_r + BN + (ReLU) + residual, fused.
__global__ void combine_kernel(const float* __restrict__ P, int ldp,
                               int hr0, int skip0,
                               const float* __restrict__ res,
                               const float* __restrict__ agg,
                               const float* __restrict__ deg,
                               const float* __restrict__ bl,
                               const float* __restrict__ g,
                               const float* __restrict__ bb,
                               const float* __restrict__ m,
                               const float* __restrict__ v,
                               float* __restrict__ hout, int relu)
{
    size_t t = (size_t)blockIdx.x * blockDim.x + threadIdx.x;
    if (t >= (size_t)NN * HH) return;
    int n = (int)(t >> 5);
    int c = (int)(t & 31);
    float mean = agg[t] / fmaxf(deg[n], 1.0f);
    float sage = mean + bl[c] + P[(size_t)n * ldp + hr0 + c];
    float bn   = (sage - m[c]) * (g[c] * rsqrtf(v[c] + EPS_)) + bb[c];
    if (relu) bn = fmaxf(bn, 0.0f);
    float r = (skip0 >= 0) ? P[(size_t)n * ldp + skip0 + c] : res[t];
    hout[t] = bn + r;
}

// One 64-thread block per graph; rank == position in stable argsort(-key).
__global__ __launch_bounds__(64)
void sortpool_kernel(const float* __restrict__ h, float* __restrict__ pooled)
{
    __shared__ float keys[NPG];
    const int gidx = blockIdx.x;
    const int t    = threadIdx.x;
    const float* hg = h + (size_t)gidx * NPG * HH;
    const float  k  = hg[t * HH + (HH - 1)];
    keys[t] = k;
    __syncthreads();
    int rank = 0;
#pragma unroll
    for (int j = 0; j < NPG; ++j) {
        float kj = keys[j];
        rank += (kj > k) || (kj == k && j < t);
    }
    if (rank < KK) {
        float* o = pooled + (size_t)gidx * (KK * HH) + rank * HH;
#pragma unroll
        for (int c = 0; c < HH; ++c) o[c] = hg[t * HH + c];
    }
}

__global__ void mlp2_kernel(const float* __restrict__ hid,
                            const float* __restrict__ w2,
                            const float* __restrict__ b2,
                            float* __restrict__ out)
{
    int gidx = blockIdx.x * blockDim.x + threadIdx.x;
    if (gidx >= G_) return;
    float s = 0.0f;
#pragma unroll
    for (int c = 0; c < HH; ++c) s += hid[(size_t)gidx * HH + c] * w2[c];
    out[gidx] = s + b2[0];
}

// ---------------------------------------------------------------------------
extern "C" void kernel_launch(void* const* d_in, const int* in_sizes, int n_in,
                              void* d_out, int out_size, void* d_ws, size_t ws_size,
                              hipStream_t stream)
{
    const float* x    = (const float*)d_in[0];
    const int*   ei   = (const int*)d_in[1];
    const int*   esrc = ei;
    const int*   edst = ei + EE;

    const float* w_l[3] = {(const float*)d_in[3],  (const float*)d_in[10], (const float*)d_in[17]};
    const float* b_l[3] = {(const float*)d_in[4],  (const float*)d_in[11], (const float*)d_in[18]};
    const float* w_r[3] = {(const float*)d_in[5],  (const float*)d_in[12], (const float*)d_in[19]};
    const float* bng[3] = {(const float*)d_in[6],  (const float*)d_in[13], (const float*)d_in[20]};
    const float* bnb[3] = {(const float*)d_in[7],  (const float*)d_in[14], (const float*)d_in[21]};
    const float* bnm[3] = {(const float*)d_in[8],  (const float*)d_in[15], (const float*)d_in[22]};
    const float* bnv[3] = {(const float*)d_in[9],  (const float*)d_in[16], (const float*)d_in[23]};
    const float* skip_w = (const float*)d_in[24];
    const float* mlp_w1 = (const float*)d_in[25];
    const float* mlp_b1 = (const float*)d_in[26];
    const float* mlp_w2 = (const float*)d_in[27];
    const float* mlp_b2 = (const float*)d_in[28];
    float* out = (float*)d_out;

    // ---- workspace carve-out (floats) ----
    float* ws     = (float*)d_ws;
    float* deg    = ws;  ws += NN;
    float* hbuf0  = ws;  ws += (size_t)NN * HH;
    float* hbuf1  = ws;  ws += (size_t)NN * HH;
    float* P      = ws;  ws += (size_t)NN * 96;
    float* agg    = ws;  ws += (size_t)NN * HH;
    float* pooled = ws;  ws += (size_t)G_ * KK * HH;
    float* hid    = ws;  ws += (size_t)G_ * HH;
    float* Wcat   = ws;  ws += (size_t)INC * 96;         // swizzled layer weights
    float* Wm1    = ws;  ws += (size_t)(KK * HH) * HH;   // swizzled mlp_w1

    const size_t NH = (size_t)NN * HH;
    const int zb_nh  = (int)((NH + 255) / 256);
    const int sc_blk = (int)(((size_t)EE * 32 + 255) / 256);
    const int gblk   = ((NN / 16) + 3) / 4;        // 1 wave per 16-row stripe

    // degree (shared by all layers)
    zero_kernel<<<(NN + 255) / 256, 256, 0, stream>>>(deg, (size_t)NN);
    degree_kernel<<<(EE + 255) / 256, 256, 0, stream>>>(edst, deg, EE);

    float* h    = hbuf0;
    float* hnew = hbuf1;

    // ---------------- layer 0: P = X * [w_l | w_r | skip]  (X read once) ---
    pack_cols_swz<<<(INC * 32 + 255) / 256, 256, 0, stream>>>(w_l[0], INC, Wcat, 96, 0);
    pack_cols_swz<<<(INC * 32 + 255) / 256, 256, 0, stream>>>(w_r[0], INC, Wcat, 96, 32);
    pack_cols_swz<<<(INC * 32 + 255) / 256, 256, 0, stream>>>(skip_w, INC, Wcat, 96, 64);
    gemm_wmma_f32<6, 128><<<gblk, 128, 0, stream>>>(x, INC, Wcat, P,
                                                    NN, INC, nullptr, 0);
    zero_kernel<<<zb_nh, 256, 0, stream>>>(agg, NH);
    scatter_add32<<<sc_blk, 256, 0, stream>>>(P, 96, esrc, edst, agg, EE);
    combine_kernel<<<zb_nh, 256, 0, stream>>>(P, 96, 32, 64, nullptr, agg, deg,
                                              b_l[0], bng[0], bnb[0], bnm[0], bnv[0],
                                              h, /*relu=*/1);

    // ---------------- layers 1, 2: P = h * [w_l | w_r]  (h read once) ------
    for (int i = 1; i < 3; ++i) {
        pack_cols_swz<<<(HH * 32 + 255) / 256, 256, 0, stream>>>(w_l[i], HH, Wcat, 64, 0);
        pack_cols_swz<<<(HH * 32 + 255) / 256, 256, 0, stream>>>(w_r[i], HH, Wcat, 64, 32);
        gemm_wmma_f32<4, 32><<<gblk, 128, 0, stream>>>(h, HH, Wcat, P,
                                                       NN, HH, nullptr, 0);
        zero_kernel<<<zb_nh, 256, 0, stream>>>(agg, NH);
        scatter_add32<<<sc_blk, 256, 0, stream>>>(P, 64, esrc, edst, agg, EE);
        combine_kernel<<<zb_nh, 256, 0, stream>>>(P, 64, 32, -1, h, agg, deg,
                                                  b_l[i], bng[i], bnb[i], bnm[i], bnv[i],
                                                  hnew, /*relu=*/(i < 2) ? 1 : 0);
        float* t = h; h = hnew; hnew = t;
    }

    // ---------------- sort-pool + MLP ----------------
    sortpool_kernel<<<G_, 64, 0, stream>>>(h, pooled);

    pack_cols_swz<<<((KK * HH) * 32 + 255) / 256, 256, 0, stream>>>(mlp_w1, KK * HH,
                                                                    Wm1, 32, 0);
    const int gblk_mlp = ((G_ / 16) + 3) / 4;
    gemm_wmma_f32<2, 96><<<gblk_mlp, 128, 0, stream>>>(pooled, KK * HH, Wm1,
                                                       hid, G_, KK * HH,
                                                       mlp_b1, /*relu=*/1);
    mlp2_kernel<<<(G_ + 255) / 256, 256, 0, stream>>>(hid, mlp_w2, mlp_b2, out);
}